// MendGraph_63771674411481
// MI455X (gfx1250) — compile-verified
//
#include <hip/hip_runtime.h>
#include <cstdint>
#include <cstddef>

#define N_ORGN 4000
#define NUM_PRED 3
#define FEAT 256
#define N_EDGES 64000
#define NODE_LEN 16000
#define NEW_EDGES (N_ORGN * NUM_PRED)              // 12000
#define TOT_EDGES (N_EDGES + NEW_EDGES)            // 76000
#define FEATS_ELEMS ((size_t)NODE_LEN * FEAT)      // 4,096,000
#define ADJ_ELEMS   ((size_t)NODE_LEN * NODE_LEN)  // 256,000,000
#define N_TILES (NODE_LEN / 64)                    // 250 WMMA tiles per row

typedef __attribute__((ext_vector_type(2))) float v2f;
typedef __attribute__((ext_vector_type(4))) float v4f;
typedef __attribute__((ext_vector_type(8))) float v8f;

// ---------------------------------------------------------------- zero fill
// Streaming 1 GB write; non-temporal so it does not thrash L2.
__global__ void zero_f4(v4f* __restrict__ p, size_t n4) {
  size_t i = (size_t)blockIdx.x * blockDim.x + threadIdx.x;
  const size_t stride = (size_t)gridDim.x * blockDim.x;
  const v4f z = {0.f, 0.f, 0.f, 0.f};
  for (; i < n4; i += stride) __builtin_nontemporal_store(z, p + i);
}

// ------------------------------------------------------- COO scatter (sums)
__global__ void scatter_edges(const int* __restrict__ edges,
                              const int* __restrict__ pred_missing,
                              float* __restrict__ adj) {
  const int tid = blockIdx.x * blockDim.x + threadIdx.x;
  if (tid < N_EDGES) {
    const int r = edges[2 * tid];
    const int c = edges[2 * tid + 1];
    atomicAdd(&adj[(size_t)r * NODE_LEN + c], 1.0f);
  } else if (tid < TOT_EDGES) {
    const int t = tid - N_EDGES;           // t = i*NUM_PRED + j
    const int i = t / NUM_PRED;
    const int j = t % NUM_PRED;
    int deg = pred_missing[i];
    deg = deg < 0 ? 0 : (deg > NUM_PRED ? NUM_PRED : deg);
    if (j < deg) {
      atomicAdd(&adj[(size_t)i * NODE_LEN + (N_ORGN + t)], 1.0f);
    }
  }
}

// --------------------------- A = max(A, A^T), only at edge-covered positions
// Race-safe: all writers of a mirror pair converge to the same max value.
__global__ void symmetrize(const int* __restrict__ edges, float* __restrict__ adj) {
  const int tid = blockIdx.x * blockDim.x + threadIdx.x;
  int r, c;
  if (tid < N_EDGES) {
    r = edges[2 * tid];
    c = edges[2 * tid + 1];
  } else if (tid < TOT_EDGES) {
    const int t = tid - N_EDGES;
    r = t / NUM_PRED;
    c = N_ORGN + t;                        // generated node column
  } else {
    return;
  }
  const size_t rc = (size_t)r * NODE_LEN + c;
  const size_t cr = (size_t)c * NODE_LEN + r;
  const float m = fmaxf(adj[rc], adj[cr]);
  adj[rc] = m;
  adj[cr] = m;
}

// ------------------------------------------------------------ A += I
__global__ void add_diag(float* __restrict__ adj) {
  const int i = blockIdx.x * blockDim.x + threadIdx.x;
  if (i < NODE_LEN) adj[(size_t)i * NODE_LEN + i] += 1.0f;
}

// --------------------------------------------- row normalize: adj = D^-1 A
// One block (8 wave32s) per row. Row (64 KB) staged global->LDS once via
// async-to-LDS b128 copies (ASYNCcnt path); rowsum folded through
// v_wmma_f32_16x16x4_f32 (B = ones => GEMV reduction, dual accumulators to
// break the D->C chain); scale pass re-reads LDS, non-temporal b128 stores.
__global__ void __launch_bounds__(256) rownorm(float* __restrict__ adj) {
  __shared__ __align__(16) float row[NODE_LEN];   // 64000 B
  __shared__ float wsum[8];
  __shared__ float rinv_s;

  const int tid = threadIdx.x;
  // wave id is wave-uniform: force scalar so the tile loop is SALU-controlled
  const int wave = __builtin_amdgcn_readfirstlane(tid >> 5);
  const int lane = tid & 31;
  float* __restrict__ g = adj + (size_t)blockIdx.x * NODE_LEN;

  // ---- stage full row into LDS with async b128 copies
  for (int i = tid; i < NODE_LEN / 4; i += 256) {
    const uint32_t lds_byte = (uint32_t)(size_t)(&row[i * 4]);
    const uint64_t gaddr = (uint64_t)(size_t)(g + i * 4);
    asm volatile("global_load_async_to_lds_b128 %0, %1, off"
                 :
                 : "v"(lds_byte), "v"(gaddr)
                 : "memory");
  }
  asm volatile("s_wait_asynccnt 0" ::: "memory");
  __syncthreads();

  // ---- rowsum
#if __has_builtin(__builtin_amdgcn_wmma_f32_16x16x4_f32)
  // Sum 64 floats per WMMA: B = ones makes D[m][n] = sum_k A[m][k], so any
  // disjoint 64-float -> 16x4 tile mapping works; lane L takes 2 consecutive
  // floats (conflict-free ds_load_b64). Two accumulators overlap load+MMA.
  v8f acc0 = {0.f, 0.f, 0.f, 0.f, 0.f, 0.f, 0.f, 0.f};
  v8f acc1 = acc0;
  v2f ones;
  ones[0] = 1.0f;
  ones[1] = 1.0f;
  int t = wave;
  for (; t + 8 < N_TILES; t += 16) {
    const v2f a0 = *(const v2f*)(&row[t * 64 + lane * 2]);
    const v2f a1 = *(const v2f*)(&row[(t + 8) * 64 + lane * 2]);
    acc0 = __builtin_amdgcn_wmma_f32_16x16x4_f32(
        false, a0, false, ones, (short)0, acc0, false, false);
    acc1 = __builtin_amdgcn_wmma_f32_16x16x4_f32(
        false, a1, false, ones, (short)0, acc1, false, false);
  }
  if (t < N_TILES) {
    const v2f a0 = *(const v2f*)(&row[t * 64 + lane * 2]);
    acc0 = __builtin_amdgcn_wmma_f32_16x16x4_f32(
        false, a0, false, ones, (short)0, acc0, false, false);
  }
  acc0 += acc1;
  // every column of D is identical; lanes 0-15 hold rows 0-7 across the 8
  // accumulator regs, lanes 16-31 hold rows 8-15.
  float s = acc0[0] + acc0[1] + acc0[2] + acc0[3] +
            acc0[4] + acc0[5] + acc0[6] + acc0[7];
  s += __shfl_xor(s, 16, 32);             // combine the two half-wave row sets
#else
  float s = 0.f;
  for (int i = tid; i < NODE_LEN / 4; i += 256) {
    const v4f v = ((const v4f*)row)[i];
    s += v[0] + v[1] + v[2] + v[3];
  }
  for (int off = 16; off > 0; off >>= 1) s += __shfl_down(s, off, 32);
#endif
  if (lane == 0) wsum[wave] = s;
  __syncthreads();
  if (tid == 0) {
    float r = 0.f;
    for (int w = 0; w < 8; ++w) r += wsum[w];
    rinv_s = (r > 0.f) ? (1.0f / r) : 0.f;
  }
  __syncthreads();

  // ---- scale from LDS, stream non-temporal b128 stores to global
  const float rv = rinv_s;
  for (int i = tid; i < NODE_LEN / 4; i += 256) {
    v4f v = ((const v4f*)row)[i];
    v *= rv;
    __builtin_nontemporal_store(v, (v4f*)g + i);
  }
}

// ---------------------------------------------------------------------------
extern "C" void kernel_launch(void* const* d_in, const int* in_sizes, int n_in,
                              void* d_out, int out_size, void* d_ws, size_t ws_size,
                              hipStream_t stream) {
  (void)in_sizes; (void)n_in; (void)out_size; (void)d_ws; (void)ws_size;

  const float* org_feats    = (const float*)d_in[0];  // [4000, 256]
  const int*   org_edges    = (const int*)  d_in[1];  // [64000, 2]
  const int*   pred_missing = (const int*)  d_in[2];  // [4000]
  const float* gen_feats    = (const float*)d_in[3];  // [4000, 768]

  float* out = (float*)d_out;
  float* adj = out + FEATS_ELEMS;

  // fill_feats = concat(org_feats, gen_feats)  (reshape is layout-identity)
  hipMemcpyAsync(out, org_feats,
                 (size_t)N_ORGN * FEAT * sizeof(float),
                 hipMemcpyDeviceToDevice, stream);
  hipMemcpyAsync(out + (size_t)N_ORGN * FEAT, gen_feats,
                 (size_t)N_ORGN * NUM_PRED * FEAT * sizeof(float),
                 hipMemcpyDeviceToDevice, stream);

  zero_f4<<<4096, 256, 0, stream>>>((v4f*)adj, ADJ_ELEMS / 4);

  const int eb = (TOT_EDGES + 255) / 256;
  scatter_edges<<<eb, 256, 0, stream>>>(org_edges, pred_missing, adj);
  symmetrize<<<eb, 256, 0, stream>>>(org_edges, adj);
  add_diag<<<(NODE_LEN + 255) / 256, 256, 0, stream>>>(adj);
  rownorm<<<NODE_LEN, 256, 0, stream>>>(adj);
}